// SelfAttentionND_10428180595056
// MI455X (gfx1250) — compile-verified
//
#include <hip/hip_runtime.h>

// ---------------------------------------------------------------------------
// Self-attention (1x1-conv QKV, softmax(q^T k) v^T, 1x1-conv out, residual)
// B=4, C=512, CI=64, N=H*W=4096.  All-fp32 via V_WMMA_F32_16X16X4_F32.
// Flash-attention streaming: the 4x4096x4096 attention matrix is never
// materialized (saves ~1 GB of memory traffic vs the naive 2-pass softmax).
// ---------------------------------------------------------------------------

#define C_IN   512
#define C_INT  64
#define NTOK   4096
#define BATCH  4

typedef __attribute__((ext_vector_type(2))) float v2f;
typedef __attribute__((ext_vector_type(8))) float v8f;

__device__ __forceinline__ v8f wmma_f32_16x16x4(v2f a, v2f b, v8f c) {
  // 8-arg pattern: (neg_a, A, neg_b, B, c_mod, C, reuse_a, reuse_b)
  return __builtin_amdgcn_wmma_f32_16x16x4_f32(false, a, false, b, (short)0, c,
                                               false, false);
}

// ---------------------------------------------------------------------------
// Weight norm (dim=0): w[o][c] = v[o][c] * g[o] / ||v[o]||.  One block per
// output row; tiny arrays, pure VALU.
// ---------------------------------------------------------------------------
__global__ void wn_kernel(const float* __restrict__ v,
                          const float* __restrict__ g,
                          float* __restrict__ w, int cin) {
  const int row = blockIdx.x;
  const int tid = threadIdx.x;
  const float* vr = v + (size_t)row * cin;
  float p = 0.f;
  for (int i = tid; i < cin; i += 64) { float t = vr[i]; p += t * t; }
  __shared__ float red[64];
  red[tid] = p;
  __syncthreads();
  for (int s = 32; s > 0; s >>= 1) {
    if (tid < s) red[tid] += red[tid + s];
    __syncthreads();
  }
  const float scale = g[row] * rsqrtf(red[0]);
  float* wr = w + (size_t)row * cin;
  for (int i = tid; i < cin; i += 64) wr[i] = vr[i] * scale;
}

// ---------------------------------------------------------------------------
// QKV projection: per wave a 16(M) x 64(N) tile of W[64,512] @ X[512,4096].
// A fragment (16x4 f32): lane m = lane%16, K pair = 2*(lane/16)+{0,1} -> b64.
// B fragment (4x16 f32): row K = 2*(lane/16)+{0,1}, col = lane%16 -> 2x b32.
// ---------------------------------------------------------------------------
__global__ __launch_bounds__(32) void qkv_kernel(
    const float* __restrict__ x,
    const float* __restrict__ wq, const float* __restrict__ wk,
    const float* __restrict__ wv,
    const float* __restrict__ qb, const float* __restrict__ kb,
    const float* __restrict__ vb,
    float* __restrict__ oq, float* __restrict__ ok, float* __restrict__ ov) {
  const int nt   = blockIdx.x;        // 0..63  : 64-wide N tile
  const int bm   = blockIdx.y;        // b*4 + mt
  const int proj = blockIdx.z;        // 0..2
  const int b = bm >> 2, mt = bm & 3;

  const float* W  = (proj == 0) ? wq : (proj == 1) ? wk : wv;
  const float* Bb = (proj == 0) ? qb : (proj == 1) ? kb : vb;
  float*       O  = ((proj == 0) ? oq : (proj == 1) ? ok : ov)
                    + (size_t)b * (C_INT * NTOK);
  const float* X  = x + (size_t)b * C_IN * NTOK;

  const int lane = threadIdx.x;
  const int lm = lane & 15, lh = lane >> 4;
  const int m0 = mt * 16, n0 = nt * 64;

  v8f acc[4] = {};
  for (int k0 = 0; k0 < C_IN; k0 += 4) {
    v2f a = *(const v2f*)(W + (m0 + lm) * C_IN + k0 + 2 * lh);
    const float* xk = X + (k0 + 2 * lh) * NTOK + n0 + lm;
#pragma unroll
    for (int j = 0; j < 4; ++j) {
      v2f bm2;
      bm2.x = xk[j * 16];
      bm2.y = xk[NTOK + j * 16];
      acc[j] = wmma_f32_16x16x4(a, bm2, acc[j]);
    }
  }
#pragma unroll
  for (int j = 0; j < 4; ++j)
#pragma unroll
    for (int r = 0; r < 8; ++r) {
      const int m = m0 + r + 8 * lh;              // D layout: M = r + 8*half
      O[m * NTOK + n0 + j * 16 + lm] = acc[j][r] + Bb[m];
    }
}

// ---------------------------------------------------------------------------
// Flash attention, one wave per (batch, 16 query rows).
//   S = Q^T K       (16 chained WMMAs over the 64-channel K dim)
//   online softmax  (row stats via __shfl_xor within 16-lane halves)
//   O += P V^T      (P re-layouted D->A through a 1 KB LDS stage)
// Output written flat as [N, CI] (== the reference's raw .view reshape).
// ---------------------------------------------------------------------------
__global__ __launch_bounds__(32) void attn_kernel(
    const float* __restrict__ q, const float* __restrict__ k,
    const float* __restrict__ v, float* __restrict__ o) {
  const int qt = blockIdx.x;          // 0..255
  const int b  = blockIdx.y;
  const float* Q = q + (size_t)b * (C_INT * NTOK);
  const float* K = k + (size_t)b * (C_INT * NTOK);
  const float* V = v + (size_t)b * (C_INT * NTOK);
  float*       O = o + (size_t)b * (NTOK * C_INT);

  const int lane = threadIdx.x;
  const int lm = lane & 15, lh = lane >> 4;
  const int n0 = qt * 16;

  // Q^T tile as 16 A-fragments (M = query row, K = channel chunk of 4)
  v2f qa[16];
#pragma unroll
  for (int c = 0; c < 16; ++c) {
    const int ch = 4 * c + 2 * lh;
    qa[c].x = Q[ch * NTOK + n0 + lm];
    qa[c].y = Q[(ch + 1) * NTOK + n0 + lm];
  }

  float mrun[8], srun[8];
#pragma unroll
  for (int r = 0; r < 8; ++r) { mrun[r] = -3.0e38f; srun[r] = 0.f; }
  v8f oacc[4] = {};

  __shared__ float pld[256];          // one 16x16 P tile

  for (int kt = 0; kt < 256; ++kt) {
    const int kn0 = kt * 16;
    v8f s = {};
#pragma unroll
    for (int c = 0; c < 16; ++c) {
      const int ch = 4 * c + 2 * lh;
      v2f bk;
      bk.x = K[ch * NTOK + kn0 + lm];
      bk.y = K[(ch + 1) * NTOK + kn0 + lm];
      s = wmma_f32_16x16x4(qa[c], bk, s);
    }
    // Online softmax: lane's VGPR r holds row (r + 8*half); the row spans the
    // 16 lanes of its half-wave, so xor masks 1,2,4,8 reduce it in place.
#pragma unroll
    for (int r = 0; r < 8; ++r) {
      float rmax = s[r];
      rmax = fmaxf(rmax, __shfl_xor(rmax, 1));
      rmax = fmaxf(rmax, __shfl_xor(rmax, 2));
      rmax = fmaxf(rmax, __shfl_xor(rmax, 4));
      rmax = fmaxf(rmax, __shfl_xor(rmax, 8));
      const float mnew  = fmaxf(mrun[r], rmax);
      const float scale = __expf(mrun[r] - mnew);
      const float p     = __expf(s[r] - mnew);
      float rsum = p;
      rsum += __shfl_xor(rsum, 1);
      rsum += __shfl_xor(rsum, 2);
      rsum += __shfl_xor(rsum, 4);
      rsum += __shfl_xor(rsum, 8);
      srun[r] = srun[r] * scale + rsum;
      mrun[r] = mnew;
      s[r] = p;
#pragma unroll
      for (int j = 0; j < 4; ++j) oacc[j][r] *= scale;
    }
    // Stage P (D layout) to LDS, re-read as A fragments. Single wave per
    // workgroup: LDS is in-order per wave; __syncthreads() is ~free and pins
    // the compiler's memory ordering.
    __syncthreads();
#pragma unroll
    for (int r = 0; r < 8; ++r) pld[(r + 8 * lh) * 16 + lm] = s[r];
    __syncthreads();
#pragma unroll
    for (int kc = 0; kc < 4; ++kc) {
      v2f pa = *(const v2f*)&pld[lm * 16 + kc * 4 + 2 * lh];
#pragma unroll
      for (int j = 0; j < 4; ++j) {
        // B fragment of V^T: K = key offset, N = channel; contiguous b64.
        v2f bv = *(const v2f*)(V + (j * 16 + lm) * NTOK + kn0 + kc * 4 + 2 * lh);
        oacc[j] = wmma_f32_16x16x4(pa, bv, oacc[j]);
      }
    }
  }
  // Normalize and store flat as [n, ci]
#pragma unroll
  for (int j = 0; j < 4; ++j)
#pragma unroll
    for (int r = 0; r < 8; ++r) {
      const int row = n0 + r + 8 * lh;
      O[row * C_INT + j * 16 + lm] = oacc[j][r] / srun[r];
    }
}

// ---------------------------------------------------------------------------
// Output 1x1 conv + bias + residual; writes both tuple outputs.
// Y is the attention output buffer read flat as [64, 4096] per batch
// (faithful to the reference's raw row-major .view reshape).
// ---------------------------------------------------------------------------
__global__ __launch_bounds__(32) void outconv_kernel(
    const float* __restrict__ x, const float* __restrict__ y,
    const float* __restrict__ wa, const float* __restrict__ ab,
    float* __restrict__ outp, float* __restrict__ resp) {
  const int pt = blockIdx.x;          // 0..63 : 64-wide spatial tile
  const int ct = blockIdx.y;          // 0..31 : 16 output channels
  const int b  = blockIdx.z;
  const float* Y = y + (size_t)b * (C_INT * NTOK);
  const int lane = threadIdx.x;
  const int lm = lane & 15, lh = lane >> 4;
  const int c0 = ct * 16, p0 = pt * 64;

  v8f acc[4] = {};
  for (int k0 = 0; k0 < C_INT; k0 += 4) {
    v2f a = *(const v2f*)(wa + (c0 + lm) * C_INT + k0 + 2 * lh);
    const float* yk = Y + (k0 + 2 * lh) * NTOK + p0 + lm;
#pragma unroll
    for (int j = 0; j < 4; ++j) {
      v2f bm2;
      bm2.x = yk[j * 16];
      bm2.y = yk[NTOK + j * 16];
      acc[j] = wmma_f32_16x16x4(a, bm2, acc[j]);
    }
  }
#pragma unroll
  for (int j = 0; j < 4; ++j)
#pragma unroll
    for (int r = 0; r < 8; ++r) {
      const int c = c0 + r + 8 * lh;
      const size_t idx = ((size_t)b * C_IN + c) * NTOK + p0 + j * 16 + lm;
      const float rv = acc[j][r] + ab[c];
      resp[idx] = rv;
      outp[idx] = x[idx] + rv;
    }
}

// ---------------------------------------------------------------------------
extern "C" void kernel_launch(void* const* d_in, const int* in_sizes, int n_in,
                              void* d_out, int out_size, void* d_ws,
                              size_t ws_size, hipStream_t stream) {
  (void)in_sizes; (void)n_in; (void)out_size; (void)ws_size;
  const float* x  = (const float*)d_in[0];
  const float* qv = (const float*)d_in[1];
  const float* qg = (const float*)d_in[2];
  const float* qb = (const float*)d_in[3];
  const float* kv = (const float*)d_in[4];
  const float* kg = (const float*)d_in[5];
  const float* kb = (const float*)d_in[6];
  const float* vv = (const float*)d_in[7];
  const float* vg = (const float*)d_in[8];
  const float* vb = (const float*)d_in[9];
  const float* av = (const float*)d_in[10];
  const float* ag = (const float*)d_in[11];
  const float* ab = (const float*)d_in[12];

  // Workspace layout (floats): normalized weights then q/k/v and pre-conv
  // attention output.  Total ~17.3 MB.
  float* ws  = (float*)d_ws;
  float* wq  = ws;                                   // 64*512
  float* wk  = ws + 32768;                           // 64*512
  float* wv_ = ws + 65536;                           // 64*512
  float* wa  = ws + 98304;                           // 512*64
  float* qbuf = ws + 131072;                         // 4*64*4096
  float* kbuf = qbuf + (size_t)BATCH * C_INT * NTOK;
  float* vbuf = kbuf + (size_t)BATCH * C_INT * NTOK;
  float* obuf = vbuf + (size_t)BATCH * C_INT * NTOK; // 4*4096*64

  wn_kernel<<<dim3(C_INT), dim3(64), 0, stream>>>(qv, qg, wq, C_IN);
  wn_kernel<<<dim3(C_INT), dim3(64), 0, stream>>>(kv, kg, wk, C_IN);
  wn_kernel<<<dim3(C_INT), dim3(64), 0, stream>>>(vv, vg, wv_, C_IN);
  wn_kernel<<<dim3(C_IN),  dim3(64), 0, stream>>>(av, ag, wa, C_INT);

  qkv_kernel<<<dim3(64, 16, 3), dim3(32), 0, stream>>>(
      x, wq, wk, wv_, qb, kb, vb, qbuf, kbuf, vbuf);

  attn_kernel<<<dim3(256, BATCH), dim3(32), 0, stream>>>(qbuf, kbuf, vbuf,
                                                         obuf);

  float* outp = (float*)d_out;
  float* resp = outp + (size_t)BATCH * C_IN * NTOK;
  outconv_kernel<<<dim3(64, 32, BATCH), dim3(32), 0, stream>>>(
      x, obuf, wa, ab, outp, resp);
}